// MonolithicExample_15994458210311
// MI455X (gfx1250) — compile-verified
//
#include <hip/hip_runtime.h>

typedef float v2f __attribute__((ext_vector_type(2)));
typedef float v8f __attribute__((ext_vector_type(8)));

#define EMBD 64
#define HIDD 64
#define NEL  120
#define MPAD 128   // 120 rows padded to 8 wave-tiles of 16

// ---------------------------------------------------------------------------
// Kernel 1: build per-element energy table with f32 WMMA.
// table[e] = silu(emb[e,:] @ W1 + b1) @ W2 + b2   for e in [0,120)
// One workgroup, 8 waves; wave w owns M-tile rows [16w, 16w+16).
// ---------------------------------------------------------------------------
__global__ __launch_bounds__(256) void energy_table_kernel(
    const float* __restrict__ emb, const float* __restrict__ W1,
    const float* __restrict__ b1,  const float* __restrict__ W2,
    const float* __restrict__ b2,  float* __restrict__ table)
{
    __shared__ float sA[MPAD][EMBD];   // 32 KB, zero-padded emb
    __shared__ float sB[EMBD][HIDD];   // 16 KB, W1
    __shared__ float sb1[HIDD];
    __shared__ float sW2[HIDD];

    const int tid = threadIdx.x;
    for (int idx = tid; idx < MPAD * EMBD; idx += 256) {
        int r = idx / EMBD, c = idx % EMBD;
        sA[r][c] = (r < NEL) ? emb[r * EMBD + c] : 0.0f;
    }
    for (int idx = tid; idx < EMBD * HIDD; idx += 256)
        sB[idx / HIDD][idx % HIDD] = W1[idx];
    if (tid < HIDD) { sb1[tid] = b1[tid]; sW2[tid] = W2[tid]; }
    __syncthreads();

    const int wave = tid >> 5;           // M tile id (0..7)
    const int lane = tid & 31;
    const int m0   = wave * 16;
    const int l16  = lane & 15;          // A: row within tile; B: col within tile
    const int koff = (lane >> 4) * 2;    // K sub-offset per ISA 32-bit A/B layout

    v8f acc[4];
    for (int nt = 0; nt < 4; ++nt)
        for (int j = 0; j < 8; ++j) acc[nt][j] = 0.0f;

    for (int k = 0; k < EMBD; k += 4) {
        v2f a;
        a.x = sA[m0 + l16][k + koff];
        a.y = sA[m0 + l16][k + koff + 1];
        for (int nt = 0; nt < 4; ++nt) {
            v2f b;
            b.x = sB[k + koff][nt * 16 + l16];
            b.y = sB[k + koff + 1][nt * 16 + l16];
            // (neg_a, A, neg_b, B, c_mod, C, reuse_a, reuse_b)
            acc[nt] = __builtin_amdgcn_wmma_f32_16x16x4_f32(
                false, a, false, b, (short)0, acc[nt], false, false);
        }
    }

    // Epilogue: silu + dot with W2, all in registers.
    // C/D layout: vgpr j holds row M = j + 8*(lane>>4); column N = nt*16 + l16.
    float partial[8];
    for (int j = 0; j < 8; ++j) partial[j] = 0.0f;
    for (int nt = 0; nt < 4; ++nt) {
        const int col = nt * 16 + l16;
        const float w2 = sW2[col];
        const float bb = sb1[col];
        for (int j = 0; j < 8; ++j) {
            float x = acc[nt][j] + bb;
            float s = x / (1.0f + __expf(-x));   // silu
            partial[j] += s * w2;
        }
    }
    // Reduce over the 16 lanes that share (lane>>4): xor offsets 1,2,4,8.
    for (int off = 1; off < 16; off <<= 1)
        for (int j = 0; j < 8; ++j)
            partial[j] += __shfl_xor(partial[j], off, 32);

    if (l16 == 0) {
        const int base = m0 + (lane >> 4) * 8;
        const float bb2 = b2[0];
        for (int j = 0; j < 8; ++j) {
            int row = base + j;
            if (row < NEL) table[row] = partial[j] + bb2;
        }
    }
}

// ---------------------------------------------------------------------------
// Kernel 2: zero the whole output buffer (forces are exactly zero; energy /
// num_atoms accumulators start at zero). float4 stores -> global_store_b128.
// ---------------------------------------------------------------------------
__global__ void zero_out_kernel(float* __restrict__ out, int n)
{
    int i4 = blockIdx.x * blockDim.x + threadIdx.x;
    int i  = i4 * 4;
    if (i + 4 <= n) {
        reinterpret_cast<float4*>(out)[i4] = make_float4(0.f, 0.f, 0.f, 0.f);
    } else if (i < n) {
        for (; i < n; ++i) out[i] = 0.0f;
    }
}

// ---------------------------------------------------------------------------
// Kernel 3: per-atom table gather + sorted-segment reduction.
// batch is sorted, so run-heads inside a block sum their run from LDS and
// issue a single atomicAdd per (run, block).
// ---------------------------------------------------------------------------
__global__ __launch_bounds__(256) void segsum_kernel(
    const int* __restrict__ z, const int* __restrict__ batch,
    const float* __restrict__ table,
    float* __restrict__ energy, float* __restrict__ num_atoms, int n)
{
    __shared__ float se[256];
    __shared__ int   sb[256];
    const int tid  = threadIdx.x;
    const int base = blockIdx.x * 256;
    const int cnt  = min(256, n - base);

    float e = 0.0f;
    int   b = -1;
    if (tid < cnt) {
        b = batch[base + tid];
        e = table[z[base + tid]];
    }
    se[tid] = e;
    sb[tid] = b;
    __syncthreads();

    if (tid < cnt) {
        const bool head = (tid == 0) || (sb[tid - 1] != b);
        if (head) {
            float sum = 0.0f, c = 0.0f;
            int j = tid;
            while (j < cnt && sb[j] == b) { sum += se[j]; c += 1.0f; ++j; }
            atomicAdd(&energy[b], sum);
            atomicAdd(&num_atoms[b], c);
        }
    }
}

// ---------------------------------------------------------------------------
extern "C" void kernel_launch(void* const* d_in, const int* in_sizes, int n_in,
                              void* d_out, int out_size, void* d_ws, size_t ws_size,
                              hipStream_t stream)
{
    const int*   z     = (const int*)d_in[0];
    // d_in[1] = pos : unused — energy is position-independent, forces == 0.
    const int*   batch = (const int*)d_in[2];
    const float* emb   = (const float*)d_in[3];
    const float* W1    = (const float*)d_in[4];
    const float* b1    = (const float*)d_in[5];
    const float* W2    = (const float*)d_in[6];
    const float* b2    = (const float*)d_in[7];

    const int N = in_sizes[0];                  // 1,000,000 atoms
    const int G = (out_size - 3 * N) / 2;       // 4096 graphs
    float* out       = (float*)d_out;
    float* energy    = out;                     // [0, G)
    float* num_atoms = out + (out_size - G);    // last G entries
    float* table     = (float*)d_ws;            // 120 floats of scratch

    // 1) tiny WMMA GEMM -> per-element energy table
    energy_table_kernel<<<1, 256, 0, stream>>>(emb, W1, b1, W2, b2, table);

    // 2) zero energy + forces + num_atoms (entire d_out)
    {
        int n4 = (out_size + 3) / 4;
        int blocks = (n4 + 255) / 256;
        zero_out_kernel<<<blocks, 256, 0, stream>>>(out, out_size);
    }

    // 3) gather + sorted segment sums
    {
        int blocks = (N + 255) / 256;
        segsum_kernel<<<blocks, 256, 0, stream>>>(z, batch, table, energy, num_atoms, N);
    }
}